// MMI_MM_Loss_36618891166334
// MI455X (gfx1250) — compile-verified
//
#include <hip/hip_runtime.h>
#include <math.h>

typedef float v2f __attribute__((ext_vector_type(2)));
typedef float v8f __attribute__((ext_vector_type(8)));

#define N_   24
#define T_   128
#define V_   32000
#define DIM_ 8

// --- CDNA5 native transcendentals (single TRANS32 ops) with safe fallbacks ---
#if __has_builtin(__builtin_amdgcn_tanhf)
  #define FAST_TANH(x) __builtin_amdgcn_tanhf(x)     // v_tanh_f32 (new on gfx1250)
#else
  #define FAST_TANH(x) tanhf(x)
#endif

#if __has_builtin(__builtin_amdgcn_logf)
  #define FAST_LOG2(x) __builtin_amdgcn_logf(x)      // v_log_f32 (log base 2)
#else
  #define FAST_LOG2(x) __log2f(x)
#endif

#if __has_builtin(__builtin_amdgcn_rcpf)
  #define FAST_RCP(x)  __builtin_amdgcn_rcpf(x)      // v_rcp_f32
#else
  #define FAST_RCP(x)  (1.0f / (x))
#endif

#define LN2_F 0.69314718055994530942f
__device__ __forceinline__ float fast_ln(float x) { return LN2_F * FAST_LOG2(x); }

__global__ __launch_bounds__(1024)
void mmi_mm_loss_kernel(const float* __restrict__ emb,
                        const int* __restrict__ targets,
                        float* __restrict__ out) {
  __shared__ float ldsLp[32];   // per-wave partial sums of log(tanh(ex))
  __shared__ float ldsLd[32];   // per-wave partial sums of log(tanh(ex/ct))
  __shared__ float ldsCt[8];    // ct_g[d, T-1]
  __shared__ float ldsLdSum[8]; // final per-d loss_div sums (from lane 16)

  const int tid  = threadIdx.x;    // 0..1023 : one thread per (d, t)
  const int d    = tid >> 7;       // 0..7
  const int t    = tid & (T_ - 1); // 0..127
  const int lane = tid & 31;
  const int wave = tid >> 5;       // 0..31 ; each wave covers a single d

  // Coalesced target fetch, then 3 scattered gathers from the 393MB tensor.
  const int tgt = targets[d * T_ + t];
  const long long rowE  = ((long long)( d            * T_ + t)) * V_ + tgt;
  const long long rowC1 = ((long long)((d +     DIM_)* T_ + t)) * V_ + tgt;
  const long long rowC2 = ((long long)((d + 2 * DIM_)* T_ + t)) * V_ + tgt;

  const float ex = emb[rowE];
  const float ct = emb[rowC1] + emb[rowC2];   // segment_sum collapses to 2-term add

  // ex in [0.1,1)  -> tanh(ex) in (0.099,0.76) : native log2/tanh are safe
  float lp = fast_ln(FAST_TANH(ex));
  float ld = fast_ln(FAST_TANH(ex * FAST_RCP(ct)));

  // wave32 butterfly reduction over the 32 t-values this wave owns
  #pragma unroll
  for (int off = 16; off >= 1; off >>= 1) {
    lp += __shfl_xor(lp, off, 32);
    ld += __shfl_xor(ld, off, 32);
  }

  if (lane == 0) { ldsLp[wave] = lp; ldsLd[wave] = ld; }
  if (t == T_ - 1) { ldsCt[d] = ct; }
  __syncthreads();

  // Collapse 4 wave-partials per d for BOTH quantities with one WMMA:
  //   A (16x4): rows 0..7  = lp partials of d=row
  //             rows 8..15 = ld partials of d=row-8
  //   B (4x16) = ones  =>  D[m,n] = sum_k A[m,k]
  // Executed by ALL waves uniformly (EXEC must be all 1s; LDS source makes
  // every wave's result identical).
  // A-fragment layout (32-bit 16x4): lanes 0-15 -> K={0,1}, lanes 16-31 -> K={2,3}.
  const int m  = lane & 15;
  const int kb = (lane >> 4) << 1;  // 0 or 2
  v2f a;
  if (m < 8) {
    a.x = ldsLp[4 * m + kb];
    a.y = ldsLp[4 * m + kb + 1];
  } else {
    a.x = ldsLd[4 * (m - 8) + kb];
    a.y = ldsLd[4 * (m - 8) + kb + 1];
  }
  v2f b; b.x = 1.0f; b.y = 1.0f;
  v8f c = {};
  // 8 args: (neg_a, A, neg_b, B, c_mod, C, reuse_a, reuse_b)
  v8f dsum = __builtin_amdgcn_wmma_f32_16x16x4_f32(
      false, a, false, b, (short)0, c, false, false);

  // D layout: VGPR r, lanes 0-15 -> D[r, lane]; lanes 16-31 -> D[8+r, lane-16]
  //   lane 0 : dsum[r] = loss_positive_sum[d=r]
  //   lane 16: dsum[r] = loss_div_sum[d=r]
  if (tid == 16) {
    #pragma unroll
    for (int r = 0; r < 8; ++r) ldsLdSum[r] = dsum[r];
  }
  __syncthreads();

  if (tid == 0) {
    float acc = 0.0f;
    #pragma unroll
    for (int r = 0; r < 8; ++r) {
      const float lpS = dsum[r];        // loss_positive[d=r]
      const float ldS = ldsLdSum[r];    // loss_div[d=r]
      const float lastTerm = FAST_TANH(fast_ln(ldsCt[r]));
      const float loss = (ldS < 0.0f) ? (-lpS - 0.05f * lastTerm) : (-lpS);
      acc += loss;
    }
    out[0] = acc * 0.125f;  // mean over DIM_=8
  }
}

extern "C" void kernel_launch(void* const* d_in, const int* in_sizes, int n_in,
                              void* d_out, int out_size, void* d_ws, size_t ws_size,
                              hipStream_t stream) {
  const float* emb     = (const float*)d_in[0];  // (24,128,32000) f32
  const int*   targets = (const int*)d_in[1];    // (8,128) i32
  float*       out     = (float*)d_out;          // scalar f32

  hipLaunchKernelGGL(mmi_mm_loss_kernel, dim3(1), dim3(1024), 0, stream,
                     emb, targets, out);
}